// Attention2_1116691497334
// MI455X (gfx1250) — compile-verified
//
#include <hip/hip_runtime.h>
#include <hip/hip_bf16.h>
#include <stddef.h>

#define BATCH 32
#define LCTX  2048
#define RNN   1024
#define AHID  512
#define MINV  (-1e8f)

typedef __attribute__((ext_vector_type(2))) float v2f;
typedef __attribute__((ext_vector_type(8))) float v8f;

// -----------------------------------------------------------------------------
// Kernel 1: att_h[32,512] = h[32,1024] @ W_h[512,1024]^T + b_h   (fp32 WMMA)
// One wave (32 threads) per 16x16 output tile. grid = (32/16)*(512/16) = 64.
// V_WMMA_F32_16X16X4_F32 fragment layouts (ISA 7.12.2, 32-bit A 16x4 / C 16x16):
//   A: lane l holds row m=(l&15), VGPR j holds K = 2*(l>>4)+j
//   B: lane l holds col n=(l&15), VGPR j holds K = 2*(l>>4)+j  (B[kk][n] = W_h[n][k+kk])
//   C: VGPR v, lanes 0-15 -> row m0+v, lanes 16-31 -> row m0+v+8; col = n0+(l&15)
// -----------------------------------------------------------------------------
__global__ void k_h2att_wmma(const float* __restrict__ h,
                             const float* __restrict__ Wh,
                             const float* __restrict__ bh,
                             float* __restrict__ att_h)
{
    const int tile = blockIdx.x;
    const int m0 = (tile & 1) * 16;          // 2 m-tiles
    const int n0 = (tile >> 1) * 16;         // 32 n-tiles
    const int lane = threadIdx.x;            // 0..31, EXEC all ones
    const int half = lane >> 4;
    const int lr   = lane & 15;

    const float* __restrict__ hrow = h  + (size_t)(m0 + lr) * RNN;   // A row
    const float* __restrict__ wrow = Wh + (size_t)(n0 + lr) * RNN;   // B col (W_h row)

    v8f c = {};
    for (int k = 0; k < RNN; k += 4) {
        const int kk = k + 2 * half;
        v2f a = *(const v2f*)(hrow + kk);
        v2f b = *(const v2f*)(wrow + kk);
        c = __builtin_amdgcn_wmma_f32_16x16x4_f32(
                /*neg_a=*/false, a, /*neg_b=*/false, b,
                /*c_mod=*/(short)0, c, /*reuse_a=*/false, /*reuse_b=*/false);
    }

    const float bias = bh[n0 + lr];
#pragma unroll
    for (int v = 0; v < 8; ++v) {
        const int mm = m0 + v + half * 8;
        att_h[(size_t)mm * AHID + n0 + lr] = c[v] + bias;
    }
}

// -----------------------------------------------------------------------------
// Kernel 2: scores[b,l] = dot(tanh(p_att[b,l,:] + att_h[b,:]), W_a) + b_a,
//           masked. Streams the 128MB p_att_feats once, coalesced.
// Block = 256 threads (8 waves), each block owns (b, 64 l's); wave-per-l.
// -----------------------------------------------------------------------------
#define LPB 64
__global__ void k_scores(const float* __restrict__ p_att,
                         const float* __restrict__ att_h,
                         const float* __restrict__ Wa,
                         const float* __restrict__ ba,
                         const unsigned char* __restrict__ mask,  // jnp bool = 1B
                         float* __restrict__ scores)
{
    __shared__ float sh[AHID];
    __shared__ float sw[AHID];

    const int b  = blockIdx.x / (LCTX / LPB);
    const int l0 = (blockIdx.x % (LCTX / LPB)) * LPB;
    const int tid = threadIdx.x;

    for (int i = tid; i < AHID; i += 256) {
        sh[i] = att_h[(size_t)b * AHID + i];
        sw[i] = Wa[i];
    }
    __syncthreads();

    const int wave = tid >> 5;
    const int lane = tid & 31;
    const float bias = ba[0];

    for (int l = l0 + wave; l < l0 + LPB; l += 8) {
        const float* __restrict__ prow = p_att + ((size_t)b * LCTX + l) * AHID;
        float acc = 0.f;
#pragma unroll
        for (int j = 0; j < AHID / 32; ++j) {
            const int idx = lane + 32 * j;          // 128B coalesced per step
            const float x = prow[idx] + sh[idx];
            acc = fmaf(tanhf(x), sw[idx], acc);
        }
#pragma unroll
        for (int off = 16; off > 0; off >>= 1)
            acc += __shfl_xor(acc, off, 32);
        if (lane == 0) {
            const size_t o = (size_t)b * LCTX + l;
            scores[o] = mask[o] ? MINV : (acc + bias);
        }
    }
}

// -----------------------------------------------------------------------------
// Kernel 3: row softmax over L=2048, one block (256 thr) per batch row.
// -----------------------------------------------------------------------------
__global__ void k_softmax(const float* __restrict__ scores,
                          float* __restrict__ weight)
{
    __shared__ float redm[8];
    __shared__ float reds[8];
    const int b = blockIdx.x, tid = threadIdx.x;
    const int lane = tid & 31, wv = tid >> 5;
    const size_t base = (size_t)b * LCTX;

    float m = -3.4e38f;
    for (int l = tid; l < LCTX; l += 256) m = fmaxf(m, scores[base + l]);
#pragma unroll
    for (int off = 16; off > 0; off >>= 1) m = fmaxf(m, __shfl_xor(m, off, 32));
    if (lane == 0) redm[wv] = m;
    __syncthreads();
    m = redm[0];
#pragma unroll
    for (int i = 1; i < 8; ++i) m = fmaxf(m, redm[i]);

    float s = 0.f;
    for (int l = tid; l < LCTX; l += 256) {
        const float e = __expf(scores[base + l] - m);
        weight[base + l] = e;
        s += e;
    }
#pragma unroll
    for (int off = 16; off > 0; off >>= 1) s += __shfl_xor(s, off, 32);
    if (lane == 0) reds[wv] = s;
    __syncthreads();
    float tot = 0.f;
#pragma unroll
    for (int i = 0; i < 8; ++i) tot += reds[i];
    const float inv = 1.0f / tot;
    for (int l = tid; l < LCTX; l += 256) weight[base + l] *= inv;
}

// -----------------------------------------------------------------------------
// Kernel 4: partial[ls,b,r] = sum over l-chunk of weight[b,l]*att_feats[b,l,r].
// Streams the 256MB att_feats once. grid = B * (R/256) * LSPLIT = 1024 blocks,
// 256 threads: thread owns one r, loop over 256 l's (weights staged in LDS).
// -----------------------------------------------------------------------------
#define LSPLIT 8
#define RCHUNK 256
__global__ void k_wsum(const float* __restrict__ att_feats,
                       const float* __restrict__ weight,
                       float* __restrict__ partial)
{
    __shared__ float wsh[LCTX / LSPLIT];
    int idx = blockIdx.x;
    const int ls = idx % LSPLIT;  idx /= LSPLIT;
    const int rc = idx % (RNN / RCHUNK);  idx /= (RNN / RCHUNK);
    const int b  = idx;
    const int tid = threadIdx.x;
    const int l0 = ls * (LCTX / LSPLIT);

    wsh[tid] = weight[(size_t)b * LCTX + l0 + tid];
    __syncthreads();

    const int r = rc * RCHUNK + tid;
    const float* __restrict__ base =
        att_feats + ((size_t)b * LCTX + l0) * RNN + r;

    float acc = 0.f;
#pragma unroll 4
    for (int i = 0; i < LCTX / LSPLIT; ++i)
        acc = fmaf(wsh[i], base[(size_t)i * RNN], acc);   // 1KB coalesced / iter

    partial[((size_t)ls * BATCH + b) * RNN + r] = acc;
}

// -----------------------------------------------------------------------------
// Kernel 5: out[b,r] = sum_ls partial[ls,b,r].   32K threads.
// -----------------------------------------------------------------------------
__global__ void k_reduce(const float* __restrict__ partial,
                         float* __restrict__ out)
{
    const int i = blockIdx.x * 256 + threadIdx.x;   // i = b*RNN + r
    float acc = 0.f;
#pragma unroll
    for (int ls = 0; ls < LSPLIT; ++ls)
        acc += partial[(size_t)ls * BATCH * RNN + i];
    out[i] = acc;
}

// -----------------------------------------------------------------------------
extern "C" void kernel_launch(void* const* d_in, const int* in_sizes, int n_in,
                              void* d_out, int out_size, void* d_ws, size_t ws_size,
                              hipStream_t stream) {
    (void)in_sizes; (void)n_in; (void)out_size; (void)ws_size;
    const float*         h      = (const float*)d_in[0];
    const float*         att_f  = (const float*)d_in[1];
    const float*         p_att  = (const float*)d_in[2];
    const unsigned char* mask   = (const unsigned char*)d_in[3];  // jnp bool: 1 byte
    const float*         Wh     = (const float*)d_in[4];
    const float*         bh     = (const float*)d_in[5];
    const float*         Wa     = (const float*)d_in[6];
    const float*         ba     = (const float*)d_in[7];
    float*               out    = (float*)d_out;

    char* ws = (char*)d_ws;
    float* att_h   = (float*)(ws);                                   //  64 KB
    float* scores  = (float*)(ws + (size_t)BATCH * AHID * 4);        // 256 KB
    float* weight  = (float*)(ws + (size_t)BATCH * (AHID + LCTX) * 4);       // 256 KB
    float* partial = (float*)(ws + (size_t)BATCH * (AHID + 2 * LCTX) * 4);   //   1 MB

    // 1. fp32 WMMA projection: 64 tiles, 1 wave each
    k_h2att_wmma<<<(BATCH / 16) * (AHID / 16), 32, 0, stream>>>(h, Wh, bh, att_h);
    // 2. tanh + alpha dot + mask (streams p_att_feats, 1024 blocks)
    k_scores<<<BATCH * (LCTX / LPB), 256, 0, stream>>>(p_att, att_h, Wa, ba, mask, scores);
    // 3. softmax per batch row
    k_softmax<<<BATCH, 256, 0, stream>>>(scores, weight);
    // 4. weighted feature sum (streams att_feats, 1024 blocks)
    k_wsum<<<BATCH * (RNN / RCHUNK) * LSPLIT, 256, 0, stream>>>(att_f, weight, partial);
    // 5. fold L-split partials
    k_reduce<<<(BATCH * RNN) / 256, 256, 0, stream>>>(partial, out);
}